// CAM_70978629533752
// MI455X (gfx1250) — compile-verified
//
#include <hip/hip_runtime.h>

typedef __attribute__((ext_vector_type(2))) float v2f;
typedef __attribute__((ext_vector_type(4))) float v4f;
typedef __attribute__((ext_vector_type(8))) float v8f;
typedef int i32x4 __attribute__((vector_size(16)));   // matches builtin's V4i

#define NBATCH 4
#define NCH    128
#define NSPAT  (48 * 48 * 48)      // 110592

#if __has_builtin(__builtin_amdgcn_global_load_async_to_lds_b128) && \
    __has_builtin(__builtin_amdgcn_s_wait_asynccnt)
#define HAVE_ASYNC_LDS 1
#else
#define HAVE_ASYNC_LDS 0
#endif

#define AS1 __attribute__((address_space(1)))
#define AS3 __attribute__((address_space(3)))

// Generic LDS pointers carry the LDS byte offset in their low 32 bits
// (flat->LDS aperture mapping truncates to addr[31:0]), so these integer
// casts are value-exact address-space conversions.
__device__ __forceinline__ void cp16_to_lds(const float* g, float* l) {
#if HAVE_ASYNC_LDS
    __builtin_amdgcn_global_load_async_to_lds_b128(
        (AS1 i32x4*)(unsigned long long)g,
        (AS3 i32x4*)(unsigned int)(unsigned long long)l,
        0, 0);
#else
    v4f v = *(const v4f*)g;
    l[0] = v.x; l[1] = v.y; l[2] = v.z; l[3] = v.w;
#endif
}

// ---------------- Kernel 1: Gram matrix  G[b] = a^T a ----------------
// 8 waves / block; wave w owns tile-row strip i=w of the 8x8 grid of 16x16
// tiles. 64-row slices of a staged in LDS (stride 136 -> the two half-waves
// of each WMMA fragment hit disjoint bank ranges), double-buffered with
// async global->LDS copies overlapping the WMMA stream.

#define GRAM_CHUNK_ROWS   64
#define GRAM_LDS_STRIDE   136
#define GRAM_ROWS_PER_BLK 1024
#define GRAM_NCHUNKS      (GRAM_ROWS_PER_BLK / GRAM_CHUNK_ROWS)   // 16
#define GRAM_GRID_X       (NSPAT / GRAM_ROWS_PER_BLK)             // 108
#define GRAM_LDS_FLOATS   (2 * GRAM_CHUNK_ROWS * GRAM_LDS_STRIDE) // 17408

__device__ __forceinline__ void gram_stage(const float* __restrict__ src,
                                           float* __restrict__ dst, int tid) {
#pragma unroll
    for (int k = 0; k < 8; ++k) {
        int p = tid + k * 256;        // 2048 float4 slots (64 x 32)
        int r = p >> 5;
        int q = p & 31;
        cp16_to_lds(src + r * NCH + q * 4, dst + r * GRAM_LDS_STRIDE + q * 4);
    }
}

__global__ __launch_bounds__(256) void cam_gram_kernel(const float* __restrict__ x,
                                                       float* __restrict__ gram)
{
    extern __shared__ float smem[];
    float* buf0 = smem;
    float* buf1 = smem + GRAM_CHUNK_ROWS * GRAM_LDS_STRIDE;

    const int b    = blockIdx.y;
    const int row0 = blockIdx.x * GRAM_ROWS_PER_BLK;
    const int tid  = threadIdx.x;
    const int wave = tid >> 5;
    const int lane = tid & 31;

    const float* xb = x + (size_t)b * NSPAT * NCH;

    v8f zero = {};
    v8f acc[8];
#pragma unroll
    for (int j = 0; j < 8; ++j) acc[j] = zero;

    const int ci   = 16 * wave + (lane & 15);
    const int koff = (lane < 16) ? 0 : 2;

    gram_stage(xb + (size_t)row0 * NCH, buf0, tid);

    for (int c = 0; c < GRAM_NCHUNKS; ++c) {
        if (c + 1 < GRAM_NCHUNKS)
            gram_stage(xb + (size_t)(row0 + (c + 1) * GRAM_CHUNK_ROWS) * NCH,
                       ((c + 1) & 1) ? buf1 : buf0, tid);
#if HAVE_ASYNC_LDS
        if (c + 1 < GRAM_NCHUNKS) __builtin_amdgcn_s_wait_asynccnt(8);
        else                      __builtin_amdgcn_s_wait_asynccnt(0);
#endif
        __syncthreads();

        const float* lds = (c & 1) ? buf1 : buf0;
#pragma unroll 4
        for (int k0 = 0; k0 < GRAM_CHUNK_ROWS; k0 += 4) {
            const float* base = &lds[(k0 + koff) * GRAM_LDS_STRIDE];
            v2f A;
            A.x = base[ci];
            A.y = base[GRAM_LDS_STRIDE + ci];
#pragma unroll
            for (int j = 0; j < 8; ++j) {
                const int dj = 16 * j + (lane & 15);
                v2f Bf;
                Bf.x = base[dj];
                Bf.y = base[GRAM_LDS_STRIDE + dj];
                acc[j] = __builtin_amdgcn_wmma_f32_16x16x4_f32(
                    false, A, false, Bf, (short)0, acc[j], false, false);
            }
        }
        __syncthreads();
    }

    // Accumulate partial Gram into global workspace.
    float* g = gram + b * NCH * NCH;
    const int rbase = 16 * wave + ((lane < 16) ? 0 : 8);
#pragma unroll
    for (int j = 0; j < 8; ++j) {
        const int d = 16 * j + (lane & 15);
#pragma unroll
        for (int r = 0; r < 8; ++r)
            atomicAdd(&g[(rbase + r) * NCH + d], acc[j][r]);
    }
}

// ---------------- Kernel 2: row softmax over 128, one wave per row ----------------
__global__ __launch_bounds__(256) void cam_softmax_kernel(float* __restrict__ gram)
{
    const int wave = threadIdx.x >> 5;
    const int lane = threadIdx.x & 31;
    const int row  = blockIdx.x * 8 + wave;      // 0 .. B*C-1 (512)
    float* p = gram + (size_t)row * NCH;

    v4f v = *(const v4f*)(p + lane * 4);
    float m = fmaxf(fmaxf(v.x, v.y), fmaxf(v.z, v.w));
#pragma unroll
    for (int off = 16; off > 0; off >>= 1) m = fmaxf(m, __shfl_xor(m, off, 32));
    v.x = __expf(v.x - m); v.y = __expf(v.y - m);
    v.z = __expf(v.z - m); v.w = __expf(v.w - m);
    float s = v.x + v.y + v.z + v.w;
#pragma unroll
    for (int off = 16; off > 0; off >>= 1) s += __shfl_xor(s, off, 32);
    const float inv = 1.0f / s;
    v.x *= inv; v.y *= inv; v.z *= inv; v.w *= inv;
    *(v4f*)(p + lane * 4) = v;
}

// ---------------- Kernel 3: out = gamma * (a @ S) + x ----------------
// S resident in LDS (stride 136); each wave double-buffers its private
// 16x128 x-tile via async global->LDS copies (stride 132 -> conflict-free
// fragment gathers), no intra-loop barriers needed.

#define S_STRIDE     136
#define X_STRIDE     132
#define OUT_SLABS    (NSPAT / 128)    // 864 slabs of 128 rows (8 waves x 16)
#define OUT_GRID_X   216
#define OUT_NITER    (OUT_SLABS / OUT_GRID_X)   // 4, exact
#define OUT_LDS_FLOATS (NCH * S_STRIDE + 8 * 2 * 16 * X_STRIDE)  // 51200

__device__ __forceinline__ void xtile_stage(const float* __restrict__ src,
                                            float* __restrict__ dst, int lane) {
#pragma unroll
    for (int r = 0; r < 16; ++r)
        cp16_to_lds(src + r * NCH + lane * 4, dst + r * X_STRIDE + lane * 4);
}

__global__ __launch_bounds__(256) void cam_out_kernel(const float* __restrict__ x,
                                                      const float* __restrict__ gram,
                                                      const float* __restrict__ gamma_p,
                                                      float* __restrict__ out)
{
    extern __shared__ float smem[];
    float* slds = smem;                       // NCH * S_STRIDE
    float* xlds = smem + NCH * S_STRIDE;      // 8 waves * 2 * 16 * X_STRIDE

    const int b    = blockIdx.y;
    const int tid  = threadIdx.x;
    const int wave = tid >> 5;
    const int lane = tid & 31;
    const float gamma = gamma_p[0];

    // Load softmax matrix S[b] (128x128) into padded LDS.
    const float* g = gram + b * NCH * NCH;
#pragma unroll
    for (int k = 0; k < 16; ++k) {
        int p = tid + k * 256;    // 4096 float4 slots
        int r = p >> 5;
        int q = p & 31;
        v4f v = *(const v4f*)(g + r * NCH + q * 4);
        float* d = &slds[r * S_STRIDE + q * 4];
        d[0] = v.x; d[1] = v.y; d[2] = v.z; d[3] = v.w;
    }

    float* xbuf0 = xlds + wave * (2 * 16 * X_STRIDE);
    float* xbuf1 = xbuf0 + 16 * X_STRIDE;
    const float* xb = x   + (size_t)b * NSPAT * NCH;
    float*       ob = out + (size_t)b * NSPAT * NCH;

    const int lrow = lane & 15;
    const int koff = (lane < 16) ? 0 : 2;
    const int rsel = (lane < 16) ? 0 : 8;
    v8f zero = {};

    // Prologue: stage first slab's x-tile for this wave.
    xtile_stage(xb + (size_t)(blockIdx.x * 128 + wave * 16) * NCH, xbuf0, lane);

    __syncthreads();   // S matrix ready for all waves

    for (int i = 0; i < OUT_NITER; ++i) {
        const int m0 = (blockIdx.x + i * OUT_GRID_X) * 128 + wave * 16;
        if (i + 1 < OUT_NITER)
            xtile_stage(xb + (size_t)((blockIdx.x + (i + 1) * OUT_GRID_X) * 128 + wave * 16) * NCH,
                        ((i + 1) & 1) ? xbuf1 : xbuf0, lane);
#if HAVE_ASYNC_LDS
        if (i + 1 < OUT_NITER) __builtin_amdgcn_s_wait_asynccnt(16);
        else                   __builtin_amdgcn_s_wait_asynccnt(0);
#endif
        const float* xw = (i & 1) ? xbuf1 : xbuf0;

        v8f acc[8];
#pragma unroll
        for (int j = 0; j < 8; ++j) acc[j] = zero;

#pragma unroll 4
        for (int k0 = 0; k0 < NCH; k0 += 4) {
            v2f A;
            A.x = xw[lrow * X_STRIDE + k0 + koff];
            A.y = xw[lrow * X_STRIDE + k0 + koff + 1];
            const float* srow = &slds[(k0 + koff) * S_STRIDE];
#pragma unroll
            for (int j = 0; j < 8; ++j) {
                const int dj = 16 * j + lrow;
                v2f Bf;
                Bf.x = srow[dj];
                Bf.y = srow[S_STRIDE + dj];
                acc[j] = __builtin_amdgcn_wmma_f32_16x16x4_f32(
                    false, A, false, Bf, (short)0, acc[j], false, false);
            }
        }

        // Epilogue: out = gamma * D + x  (row-contiguous 16-lane stores).
        // ds_load -> global_store data dependence retires these LDS reads
        // before the next iteration's async overwrite is issued.
#pragma unroll
        for (int j = 0; j < 8; ++j) {
            const int col = 16 * j + lrow;
#pragma unroll
            for (int r = 0; r < 8; ++r) {
                const int lr = rsel + r;
                const float xv = xw[lr * X_STRIDE + col];
                ob[(size_t)(m0 + lr) * NCH + col] = gamma * acc[j][r] + xv;
            }
        }
    }
}

extern "C" void kernel_launch(void* const* d_in, const int* in_sizes, int n_in,
                              void* d_out, int out_size, void* d_ws, size_t ws_size,
                              hipStream_t stream)
{
    const float* x     = (const float*)d_in[0];
    const float* gamma = (const float*)d_in[1];
    float*       out   = (float*)d_out;
    float*       gram  = (float*)d_ws;          // 4*128*128 f32 = 256 KB

    (void)in_sizes; (void)n_in; (void)out_size; (void)ws_size;

    const size_t gram_shmem = (size_t)GRAM_LDS_FLOATS * sizeof(float);   // 69632 B
    const size_t out_shmem  = (size_t)OUT_LDS_FLOATS * sizeof(float);    // 204800 B
    (void)hipFuncSetAttribute((const void*)cam_gram_kernel,
                              hipFuncAttributeMaxDynamicSharedMemorySize, (int)gram_shmem);
    (void)hipFuncSetAttribute((const void*)cam_out_kernel,
                              hipFuncAttributeMaxDynamicSharedMemorySize, (int)out_shmem);

    (void)hipMemsetAsync(gram, 0, (size_t)NBATCH * NCH * NCH * sizeof(float), stream);

    cam_gram_kernel<<<dim3(GRAM_GRID_X, NBATCH), 256, gram_shmem, stream>>>(x, gram);
    cam_softmax_kernel<<<dim3((NBATCH * NCH) / 8), 256, 0, stream>>>(gram);
    cam_out_kernel<<<dim3(OUT_GRID_X, NBATCH), 256, out_shmem, stream>>>(x, gram, gamma, out);
}